// MemNN_88046829568565
// MI455X (gfx1250) — compile-verified
//
#include <hip/hip_runtime.h>
#include <hip/hip_bf16.h>

#define HOPS   3
#define VOCAB  50000
#define EMBD   128
#define SLEN   200
#define SENT   32
#define BATCH  64
#define QLEN   32
#define NCHUNK 25   // 200 sentences / 8 per chunk

typedef __attribute__((ext_vector_type(2))) float v2f;
typedef __attribute__((ext_vector_type(8))) float v8f;

// ---------------------------------------------------------------------------
// u[b][e] = sum_j A1[q[b][j]][e]
// ---------------------------------------------------------------------------
__global__ __launch_bounds__(EMBD)
void k_init_u(const float* __restrict__ A1, const int* __restrict__ q,
              float* __restrict__ u) {
    int b = blockIdx.x, e = threadIdx.x;
    const int* qb = q + b * QLEN;
    float acc = 0.f;
#pragma unroll 4
    for (int j = 0; j < QLEN; ++j)
        acc += A1[(long)qb[j] * EMBD + e];
    u[b * EMBD + e] = acc;
}

// ---------------------------------------------------------------------------
// praw[b][s] = sum_e (sum_t Ak[x[b][s][t]][e] + TA[s]) * u[b][e]
// One block per (s, b); thread = embedding dim; coalesced 512B gathers.
// ---------------------------------------------------------------------------
__global__ __launch_bounds__(EMBD)
void k_score(const float* __restrict__ Ak, const float* __restrict__ TA,
             const int* __restrict__ x, const float* __restrict__ u,
             float* __restrict__ praw) {
    int s = blockIdx.x, b = blockIdx.y, e = threadIdx.x;
    const int* xs = x + (b * SLEN + s) * SENT;
    float acc = 0.f;
#pragma unroll 4
    for (int t = 0; t < SENT; ++t)
        acc += Ak[(long)xs[t] * EMBD + e];
    acc += TA[s];
    float v = acc * u[b * EMBD + e];

    __shared__ float red[EMBD];
    red[e] = v;
    __syncthreads();
    for (int off = EMBD / 2; off > 0; off >>= 1) {
        if (e < off) red[e] += red[e + off];
        __syncthreads();
    }
    if (e == 0) praw[b * SLEN + s] = red[0];
}

// ---------------------------------------------------------------------------
// p[b][:] = softmax(p[b][:]) over S=200, in place.
// ---------------------------------------------------------------------------
__global__ __launch_bounds__(256)
void k_softmax(float* __restrict__ p) {
    int b = blockIdx.x, t = threadIdx.x;
    float* pb = p + b * SLEN;
    __shared__ float red[256];

    float m = -1e30f;
    for (int s = t; s < SLEN; s += 256) m = fmaxf(m, pb[s]);
    red[t] = m;
    __syncthreads();
    for (int off = 128; off > 0; off >>= 1) {
        if (t < off) red[t] = fmaxf(red[t], red[t + off]);
        __syncthreads();
    }
    m = red[0];
    __syncthreads();

    float sum = 0.f;
    for (int s = t; s < SLEN; s += 256) sum += __expf(pb[s] - m);
    red[t] = sum;
    __syncthreads();
    for (int off = 128; off > 0; off >>= 1) {
        if (t < off) red[t] += red[t + off];
        __syncthreads();
    }
    float inv = 1.0f / red[0];
    for (int s = t; s < SLEN; s += 256) pb[s] = __expf(pb[s] - m) * inv;
}

// ---------------------------------------------------------------------------
// part[b][ch][e] = sum_{s in chunk} p[b][s] * (sum_t Ak1[x[b][s][t]][e]+TC[s])
// Deterministic two-stage reduction (no float atomics).
// ---------------------------------------------------------------------------
__global__ __launch_bounds__(EMBD)
void k_combine(const float* __restrict__ Ak1, const float* __restrict__ TC,
               const int* __restrict__ x, const float* __restrict__ p,
               float* __restrict__ part) {
    int ch = blockIdx.x, b = blockIdx.y, e = threadIdx.x;
    float acc = 0.f;
    for (int s = ch * 8; s < ch * 8 + 8; ++s) {
        const int* xs = x + (b * SLEN + s) * SENT;
        float g = 0.f;
#pragma unroll 4
        for (int t = 0; t < SENT; ++t)
            g += Ak1[(long)xs[t] * EMBD + e];
        acc += p[b * SLEN + s] * (g + TC[s]);
    }
    part[(b * NCHUNK + ch) * EMBD + e] = acc;
}

// u[b][e] += sum_ch part[b][ch][e]   (residual)
__global__ __launch_bounds__(EMBD)
void k_update(float* __restrict__ u, const float* __restrict__ part) {
    int b = blockIdx.x, e = threadIdx.x;
    float acc = 0.f;
    for (int ch = 0; ch < NCHUNK; ++ch)
        acc += part[(b * NCHUNK + ch) * EMBD + e];
    u[b * EMBD + e] += acc;
}

// ---------------------------------------------------------------------------
// logits[m][n] = sum_k u[m][k] * A3[n][k]   via V_WMMA_F32_16X16X4_F32.
// One wave32 per 16-wide vocab column strip, computing ALL 4 batch tiles
// (M=64) so each B fragment (A3 tile) is loaded once and reused 4x.
// f32 A-frag layout: V0 = K0 (lanes 0-15) | K2 (lanes 16-31); V1 = K1 | K3.
// B (4x16, = A3 tile transposed) mirrors that with lane = N column.
// C/D: VGPR r holds row m0 + r + 8*(lane>=16), col = lane & 15.
// ---------------------------------------------------------------------------
__global__ __launch_bounds__(32)
void k_logits(const float* __restrict__ u, const float* __restrict__ A3,
              float* __restrict__ out) {
    int n0   = blockIdx.x * 16;          // vocab tile
    int lane = threadIdx.x;
    int half = lane >> 4;                // 0: K even-pair, 1: K odd-pair
    int l16  = lane & 15;
    int koff = half * 2;

    const float* arow = A3 + (long)(n0 + l16) * EMBD;   // B-frag source
    const float* u0   = u + l16 * EMBD;                 // m-tile 0 row base

    v8f c0 = {}, c1 = {}, c2 = {}, c3 = {};
#pragma unroll
    for (int k0 = 0; k0 < EMBD; k0 += 4) {
        int kk = k0 + koff;
        v2f bfrag, a0, a1, a2, a3;
        bfrag.x = arow[kk];               bfrag.y = arow[kk + 1];
        a0.x = u0[kk];                    a0.y = u0[kk + 1];
        a1.x = u0[16 * EMBD + kk];        a1.y = u0[16 * EMBD + kk + 1];
        a2.x = u0[32 * EMBD + kk];        a2.y = u0[32 * EMBD + kk + 1];
        a3.x = u0[48 * EMBD + kk];        a3.y = u0[48 * EMBD + kk + 1];
        c0 = __builtin_amdgcn_wmma_f32_16x16x4_f32(false, a0, false, bfrag,
                                                   (short)0, c0, false, false);
        c1 = __builtin_amdgcn_wmma_f32_16x16x4_f32(false, a1, false, bfrag,
                                                   (short)0, c1, false, false);
        c2 = __builtin_amdgcn_wmma_f32_16x16x4_f32(false, a2, false, bfrag,
                                                   (short)0, c2, false, false);
        c3 = __builtin_amdgcn_wmma_f32_16x16x4_f32(false, a3, false, bfrag,
                                                   (short)0, c3, false, false);
    }

    int col = n0 + l16;
#pragma unroll
    for (int r = 0; r < 8; ++r) {
        int row = r + 8 * half;
        out[(long)(row)      * VOCAB + col] = c0[r];
        out[(long)(row + 16) * VOCAB + col] = c1[r];
        out[(long)(row + 32) * VOCAB + col] = c2[r];
        out[(long)(row + 48) * VOCAB + col] = c3[r];
    }
}

// ---------------------------------------------------------------------------
// Per-row max and log-sum-exp over V=50000.
// ---------------------------------------------------------------------------
__global__ __launch_bounds__(256)
void k_rowstats(const float* __restrict__ out, float* __restrict__ rowmax,
                float* __restrict__ rowlse) {
    int b = blockIdx.x, t = threadIdx.x;
    const float* row = out + (long)b * VOCAB;
    __shared__ float red[256];

    float m = -1e30f;
    for (int v = t; v < VOCAB; v += 256) m = fmaxf(m, row[v]);
    red[t] = m;
    __syncthreads();
    for (int off = 128; off > 0; off >>= 1) {
        if (t < off) red[t] = fmaxf(red[t], red[t + off]);
        __syncthreads();
    }
    m = red[0];
    __syncthreads();

    float s = 0.f;
    for (int v = t; v < VOCAB; v += 256) s += __expf(row[v] - m);
    red[t] = s;
    __syncthreads();
    for (int off = 128; off > 0; off >>= 1) {
        if (t < off) red[t] += red[t + off];
        __syncthreads();
    }
    if (t == 0) { rowmax[b] = m; rowlse[b] = __logf(red[0]); }
}

// out[i] = out[i] - rowmax[b] - rowlse[b]
__global__ __launch_bounds__(256)
void k_finalize(float* __restrict__ out, const float* __restrict__ rowmax,
                const float* __restrict__ rowlse) {
    long i = (long)blockIdx.x * 256 + threadIdx.x;
    if (i < (long)BATCH * VOCAB) {
        int b = (int)(i / VOCAB);
        out[i] = out[i] - rowmax[b] - rowlse[b];
    }
}

// ---------------------------------------------------------------------------
extern "C" void kernel_launch(void* const* d_in, const int* in_sizes, int n_in,
                              void* d_out, int out_size, void* d_ws, size_t ws_size,
                              hipStream_t stream) {
    const float* A  = (const float*)d_in[0];   // [4, V, E]
    const float* TA = (const float*)d_in[1];   // [S]
    const float* TC = (const float*)d_in[2];   // [S]
    const int*   x  = (const int*)d_in[3];     // [B, S, T]
    const int*   q  = (const int*)d_in[4];     // [B, Q]
    float* out = (float*)d_out;                // [B, V]

    float* ws   = (float*)d_ws;
    float* u    = ws;                             // B*E       = 8192
    float* p    = u    + BATCH * EMBD;            // B*S       = 12800
    float* part = p    + BATCH * SLEN;            // B*25*E    = 204800
    float* rmax = part + BATCH * NCHUNK * EMBD;   // B
    float* rlse = rmax + BATCH;                   // B

    const long TBL = (long)VOCAB * EMBD;

    k_init_u<<<BATCH, EMBD, 0, stream>>>(A + 1 * TBL, q, u);

    for (int k = 0; k < HOPS; ++k) {
        k_score  <<<dim3(SLEN, BATCH),   EMBD, 0, stream>>>(A + k * TBL, TA, x, u, p);
        k_softmax<<<BATCH,               256,  0, stream>>>(p);
        k_combine<<<dim3(NCHUNK, BATCH), EMBD, 0, stream>>>(A + (k + 1) * TBL, TC, x, p, part);
        k_update <<<BATCH,               EMBD, 0, stream>>>(u, part);
    }

    k_logits  <<<dim3(VOCAB / 16), 32, 0, stream>>>(u, A + HOPS * TBL, out);
    k_rowstats<<<BATCH, 256, 0, stream>>>(out, rmax, rlse);
    k_finalize<<<(BATCH * VOCAB + 255) / 256, 256, 0, stream>>>(out, rmax, rlse);

    (void)in_sizes; (void)n_in; (void)out_size; (void)ws_size;
}